// GraphSAGERiskPredictor_15281493639283
// MI455X (gfx1250) — compile-verified
//
#include <hip/hip_runtime.h>

#define N_NODES 50000
#define N_EDGES 800000
#define DIM 128
#define N_TILES (N_NODES / 16)   // 3125, exact

typedef __attribute__((ext_vector_type(2))) float v2f;
typedef __attribute__((ext_vector_type(8))) float v8f;

__device__ __forceinline__ void atomic_add_f32(float* p, float v) {
    (void)__hip_atomic_fetch_add(p, v, __ATOMIC_RELAXED, __HIP_MEMORY_SCOPE_AGENT);
}

// ---------------- utility kernels ----------------

__global__ void k_fill0(float4* __restrict__ p, int n4) {
    int i = blockIdx.x * blockDim.x + threadIdx.x;
    if (i < n4) p[i] = make_float4(0.f, 0.f, 0.f, 0.f);
}

__global__ void k_count(const int* __restrict__ dst, float* __restrict__ cnt, int nE) {
    int e = blockIdx.x * blockDim.x + threadIdx.x;
    if (e < nE) atomic_add_f32(&cnt[dst[e]], 1.0f);
}

__global__ void k_invert(float* __restrict__ c, int n) {
    int i = blockIdx.x * blockDim.x + threadIdx.x;
    if (i < n) c[i] = 1.0f / fmaxf(c[i], 1.0f);
}

// one wave per edge; lane covers 4 features -> coalesced b128 gather + f32 atomics
__global__ void k_aggregate(const float* __restrict__ X, const int* __restrict__ src,
                            const int* __restrict__ dst, float* __restrict__ agg, int nE) {
    int t = blockIdx.x * blockDim.x + threadIdx.x;
    int e = t >> 5;
    int lane = t & 31;
    if (e >= nE) return;
    int s = src[e];
    int d = dst[e];
    const float4 v = *(const float4*)(X + (size_t)s * DIM + lane * 4);
    float* p = agg + (size_t)d * DIM + lane * 4;
    atomic_add_f32(p + 0, v.x);
    atomic_add_f32(p + 1, v.y);
    atomic_add_f32(p + 2, v.z);
    atomic_add_f32(p + 3, v.w);
}

// ---------------- SAGE conv GEMM: OUT = X@Wl^T + bl + (AGG*inv)@Wr^T ----------------
// One wave per 16-row tile; 8 live 16x16 f32 accumulators (full 128-wide output).
// A layout (16x4 f32): lane holds row M=lane&15; element j <-> K = (lane>>4)*2 + j.
// B layout (4x16 f32): lane holds col N=lane&15; element j <-> K = (lane>>4)*2 + j.
// C/D layout: element j <-> row M = (lane>>4)*8 + j, col N = lane&15.
__global__ __launch_bounds__(256) void k_sage_gemm(
    const float* __restrict__ X, const float* __restrict__ AGG,
    const float* __restrict__ INV,
    const float* __restrict__ Wl, const float* __restrict__ bl,
    const float* __restrict__ Wr, float* __restrict__ OUT, int relu)
{
    int wave = blockIdx.x * (blockDim.x >> 5) + (threadIdx.x >> 5);
    if (wave >= N_TILES) return;               // wave-uniform: EXEC stays all-ones
    int lane = threadIdx.x & 31;
    int m    = lane & 15;
    int koff = (lane >> 4) * 2;
    int row  = wave * 16 + m;
    float inv = INV[row];
    const float* xr = X   + (size_t)row * DIM;
    const float* ar = AGG + (size_t)row * DIM;

    v8f zero = {0.f, 0.f, 0.f, 0.f, 0.f, 0.f, 0.f, 0.f};
    v8f acc[8];
#pragma unroll
    for (int nt = 0; nt < 8; ++nt) acc[nt] = zero;

#pragma unroll 4
    for (int s = 0; s < 32; ++s) {
        int kk = s * 4 + koff;
        v2f ax = { xr[kk], xr[kk + 1] };
        v2f am = { ar[kk] * inv, ar[kk + 1] * inv };
#pragma unroll
        for (int nt = 0; nt < 8; ++nt) {
            int n = nt * 16 + m;
            const float* wl = Wl + (size_t)n * DIM + kk;
            const float* wr = Wr + (size_t)n * DIM + kk;
            v2f bL = { wl[0], wl[1] };
            v2f bR = { wr[0], wr[1] };
            acc[nt] = __builtin_amdgcn_wmma_f32_16x16x4_f32(
                false, ax, false, bL, (short)0, acc[nt], false, false);
            acc[nt] = __builtin_amdgcn_wmma_f32_16x16x4_f32(
                false, am, false, bR, (short)0, acc[nt], false, false);
        }
    }

    int m0 = wave * 16 + (lane >> 4) * 8;
#pragma unroll
    for (int nt = 0; nt < 8; ++nt) {
        int n = nt * 16 + m;
        float bias = bl[n];
#pragma unroll
        for (int j = 0; j < 8; ++j) {
            float v = acc[nt][j] + bias;
            if (relu) v = fmaxf(v, 0.f);
            OUT[(size_t)(m0 + j) * DIM + n] = v;
        }
    }
}

// ---------------- classifier hidden: HID = relu(EMB @ Wc1^T + bc1), 128 -> 64 ----------------
__global__ __launch_bounds__(256) void k_cls_hidden(
    const float* __restrict__ EMB, const float* __restrict__ Wc1,
    const float* __restrict__ bc1, float* __restrict__ HID)
{
    int wave = blockIdx.x * (blockDim.x >> 5) + (threadIdx.x >> 5);
    if (wave >= N_TILES) return;
    int lane = threadIdx.x & 31;
    int m    = lane & 15;
    int koff = (lane >> 4) * 2;
    int row  = wave * 16 + m;
    const float* er = EMB + (size_t)row * DIM;

    v8f zero = {0.f, 0.f, 0.f, 0.f, 0.f, 0.f, 0.f, 0.f};
    v8f acc[4];
#pragma unroll
    for (int nt = 0; nt < 4; ++nt) acc[nt] = zero;

#pragma unroll 4
    for (int s = 0; s < 32; ++s) {
        int kk = s * 4 + koff;
        v2f a = { er[kk], er[kk + 1] };
#pragma unroll
        for (int nt = 0; nt < 4; ++nt) {
            int n = nt * 16 + m;
            const float* w = Wc1 + (size_t)n * DIM + kk;
            v2f b = { w[0], w[1] };
            acc[nt] = __builtin_amdgcn_wmma_f32_16x16x4_f32(
                false, a, false, b, (short)0, acc[nt], false, false);
        }
    }

    int m0 = wave * 16 + (lane >> 4) * 8;
#pragma unroll
    for (int nt = 0; nt < 4; ++nt) {
        int n = nt * 16 + m;
        float bias = bc1[n];
#pragma unroll
        for (int j = 0; j < 8; ++j) {
            float v = fmaxf(acc[nt][j] + bias, 0.f);
            HID[(size_t)(m0 + j) * 64 + n] = v;
        }
    }
}

// ---------------- classifier out: OUT = HID @ Wc2^T + bc2, 64 -> 2 ----------------
__global__ void k_cls_out(const float* __restrict__ HID, const float* __restrict__ Wc2,
                          const float* __restrict__ bc2, float* __restrict__ OUT) {
    int t = blockIdx.x * blockDim.x + threadIdx.x;
    int node = t >> 1;
    int o = t & 1;
    if (node >= N_NODES) return;
    const float* h = HID + (size_t)node * 64;
    const float* w = Wc2 + o * 64;
    float acc = bc2[o];
#pragma unroll 8
    for (int j = 0; j < 64; ++j) acc = fmaf(h[j], w[j], acc);
    OUT[node * 2 + o] = acc;
}

// ---------------- launch ----------------

extern "C" void kernel_launch(void* const* d_in, const int* in_sizes, int n_in,
                              void* d_out, int out_size, void* d_ws, size_t ws_size,
                              hipStream_t stream) {
    (void)in_sizes; (void)n_in; (void)out_size; (void)ws_size;

    const float* x   = (const float*)d_in[0];
    const int*   ei  = (const int*)d_in[1];
    const float* W1l = (const float*)d_in[2];
    const float* b1l = (const float*)d_in[3];
    const float* W1r = (const float*)d_in[4];
    const float* W2l = (const float*)d_in[5];
    const float* b2l = (const float*)d_in[6];
    const float* W2r = (const float*)d_in[7];
    const float* W3l = (const float*)d_in[8];
    const float* b3l = (const float*)d_in[9];
    const float* W3r = (const float*)d_in[10];
    const float* Wc1 = (const float*)d_in[11];
    const float* bc1 = (const float*)d_in[12];
    const float* Wc2 = (const float*)d_in[13];
    const float* bc2 = (const float*)d_in[14];

    const int* src = ei;
    const int* dst = ei + N_EDGES;

    float* ws  = (float*)d_ws;
    float* cnt = ws;                                 // N (holds inv-degree after k_invert)
    float* agg = cnt + N_NODES;                      // N*128
    float* h1  = agg + (size_t)N_NODES * DIM;        // N*128
    float* h2  = h1  + (size_t)N_NODES * DIM;        // N*128
    float* hid = agg;                                // reuse agg after layer 3 (N*64)

    float* out = (float*)d_out;                      // [N, 2]
    float* emb = out + (size_t)N_NODES * 2;          // [N, 128]

    const int TB = 256;
    const int gridNodes  = (N_NODES + TB - 1) / TB;
    const int gridEdges  = (N_EDGES + TB - 1) / TB;
    const int gridAgg    = (N_EDGES * 32 + TB - 1) / TB;   // one wave per edge
    const int gridGemm   = (N_TILES + 7) / 8;              // 8 waves per block
    const int aggF4      = N_NODES * DIM / 4;
    const int gridAggZ   = (aggF4 + TB - 1) / TB;

    // inverse mean-degree (edges constant across layers)
    k_fill0<<<(N_NODES / 4 + TB - 1) / TB, TB, 0, stream>>>((float4*)cnt, N_NODES / 4);
    k_count<<<gridEdges, TB, 0, stream>>>(dst, cnt, N_EDGES);
    k_invert<<<gridNodes, TB, 0, stream>>>(cnt, N_NODES);

    // layer 1: x -> h1 (relu)
    k_fill0<<<gridAggZ, TB, 0, stream>>>((float4*)agg, aggF4);
    k_aggregate<<<gridAgg, TB, 0, stream>>>(x, src, dst, agg, N_EDGES);
    k_sage_gemm<<<gridGemm, TB, 0, stream>>>(x, agg, cnt, W1l, b1l, W1r, h1, 1);

    // layer 2: h1 -> h2 (relu)
    k_fill0<<<gridAggZ, TB, 0, stream>>>((float4*)agg, aggF4);
    k_aggregate<<<gridAgg, TB, 0, stream>>>(h1, src, dst, agg, N_EDGES);
    k_sage_gemm<<<gridGemm, TB, 0, stream>>>(h1, agg, cnt, W2l, b2l, W2r, h2, 1);

    // layer 3: h2 -> emb (no relu), written straight into d_out
    k_fill0<<<gridAggZ, TB, 0, stream>>>((float4*)agg, aggF4);
    k_aggregate<<<gridAgg, TB, 0, stream>>>(h2, src, dst, agg, N_EDGES);
    k_sage_gemm<<<gridGemm, TB, 0, stream>>>(h2, agg, cnt, W3l, b3l, W3r, emb, 0);

    // classifier
    k_cls_hidden<<<gridGemm, TB, 0, stream>>>(emb, Wc1, bc1, hid);
    k_cls_out<<<(N_NODES * 2 + TB - 1) / TB, TB, 0, stream>>>(hid, Wc2, bc2, out);
}